// ArrivalTime_53635551592486
// MI455X (gfx1250) — compile-verified
//
#include <hip/hip_runtime.h>

typedef float v2f __attribute__((ext_vector_type(2)));
typedef float v8f __attribute__((ext_vector_type(8)));

#define NH   4
#define HD   256
#define DM   1024
#define D2   2048
#define TT   24
#define TP   32
#define BROWS 32          // rows per block (2 row-tiles of 16)
#define NTHREADS 512      // 16 waves; each wave owns 64 output columns

__device__ __forceinline__ v8f wmma_f32(v2f a, v2f b, v8f c) {
    // V_WMMA_F32_16X16X4_F32 : D = A(16x4) x B(4x16) + C(16x16), all fp32
    return __builtin_amdgcn_wmma_f32_16x16x4_f32(
        false, a, false, b, (short)0, c, false, false);
}

// ---------------------------------------------------------------------------
// Kernel 1: k,v = emb[T,D] x Wk/Wv^T + b  ->  kbuf/vbuf [NH][TP][HD] (T-padded)
// ---------------------------------------------------------------------------
__global__ __launch_bounds__(256) void kv_kernel(
    const float* __restrict__ emb,
    const float* __restrict__ Wk, const float* __restrict__ bk,
    const float* __restrict__ Wv, const float* __restrict__ bv,
    float* __restrict__ kbuf, float* __restrict__ vbuf)
{
    __shared__ float erow[DM];
    const int h  = blockIdx.x >> 5;
    const int tp = blockIdx.x & 31;
    const int c  = threadIdx.x;
    const int o  = (h * TP + tp) * HD + c;
    if (tp >= TT) { kbuf[o] = 0.f; vbuf[o] = 0.f; return; }
    for (int i = c; i < DM; i += 256) erow[i] = emb[tp * DM + i];
    __syncthreads();
    const float* wk = Wk + (size_t)(h * HD + c) * DM;
    const float* wv = Wv + (size_t)(h * HD + c) * DM;
    float sk = 0.f, sv = 0.f;
    for (int d = 0; d < DM; ++d) {
        float e = erow[d];
        sk = fmaf(wk[d], e, sk);
        sv = fmaf(wv[d], e, sv);
    }
    kbuf[o] = sk + bk[h * HD + c];
    vbuf[o] = sv + bv[h * HD + c];
}

// ---------------------------------------------------------------------------
// Kernel 2: fully fused  gather -> GEMM1 -> attention(softmax) -> GEMM2
// One block = 32 query rows, 16 waves; wave owns 64 cols x 32 rows.
// Each B fragment (global_load_b64) feeds TWO wmmas (row-tiles 0 and 1).
// ---------------------------------------------------------------------------
__global__ __launch_bounds__(NTHREADS) void fused_kernel(
    const float* __restrict__ emb,
    const int*   __restrict__ user_x1,
    const int*   __restrict__ hour_x1,
    const int*   __restrict__ hour_mask,
    const float* __restrict__ up,
    const float* __restrict__ Wq, const float* __restrict__ bq,
    const float* __restrict__ kbuf, const float* __restrict__ vbuf,
    const float* __restrict__ Wu, const float* __restrict__ bu,
    float* __restrict__ out)
{
    __shared__ float chunkA[32 * 68];     // 32 x 64 A-chunk (+4 pad)  ~8.7 KB
    __shared__ float qtile[32 * 1028];    // q tile, reused as attn-out ~131.6 KB
    __shared__ float sc[NH * 32 * TP];    // scores -> probabilities    ~16 KB
    __shared__ int   hrs[32];
    __shared__ int   uidx;

    const int tid  = threadIdx.x;
    const int lane = tid & 31;
    const int wave = tid >> 5;            // 0..15
    const int r0   = blockIdx.x * BROWS;

    if (tid < 32)  hrs[tid] = hour_x1[r0 + tid];
    if (tid == 0)  uidx = user_x1[r0 >> 9];     // batch row constant per block
    __syncthreads();

    const int   u    = uidx;                    // padding_idx = 0 -> zeros
    const float* urow = up + (size_t)u * DM;

    const int ln    = lane & 15;                // N / M position within tile
    const int koff  = (lane < 16) ? 0 : 2;      // K half-split for A/B frags
    const int rhalf = (lane < 16) ? 0 : 8;      // row half-split for C frags
    const int cb    = wave * 64;                // this wave's output columns

    // ===================== GEMM1: q = query[32x2048] x Wq^T ================
    v8f acc[2][4];
#pragma unroll
    for (int rt = 0; rt < 2; ++rt)
#pragma unroll
        for (int j = 0; j < 4; ++j)
            acc[rt][j] = {0.f,0.f,0.f,0.f,0.f,0.f,0.f,0.f};

    for (int kk = 0; kk < D2; kk += 64) {
        __syncthreads();
        // cooperative gather of query chunk [32 rows x 64 k]
#pragma unroll
        for (int i = 0; i < 4; ++i) {
            int e   = tid + i * NTHREADS;
            int row = e >> 6, col = e & 63;
            int gc  = kk + col;
            float v;
            if (gc < DM) v = (u == 0) ? 0.f : urow[gc];
            else         v = emb[hrs[row] * DM + (gc - DM)];
            chunkA[row * 68 + col] = v;
        }
        __syncthreads();
#pragma unroll
        for (int ks = 0; ks < 64; ks += 4) {
            const float2 a0 = *(const float2*)&chunkA[ln * 68 + ks + koff];
            const float2 a1 = *(const float2*)&chunkA[(16 + ln) * 68 + ks + koff];
            v2f af0 = {a0.x, a0.y};
            v2f af1 = {a1.x, a1.y};
#pragma unroll
            for (int j = 0; j < 4; ++j) {
                int col = cb + j * 16 + ln;
                const float2 b2 =
                    *(const float2*)&Wq[(size_t)col * D2 + kk + ks + koff];
                v2f bf = {b2.x, b2.y};
                acc[0][j] = wmma_f32(af0, bf, acc[0][j]);
                acc[1][j] = wmma_f32(af1, bf, acc[1][j]);
            }
        }
    }
    // q + bias -> LDS
#pragma unroll
    for (int j = 0; j < 4; ++j) {
        int col = cb + j * 16 + ln;
        float bias = bq[col];
#pragma unroll
        for (int rt = 0; rt < 2; ++rt)
#pragma unroll
            for (int v = 0; v < 8; ++v)
                qtile[(rt * 16 + v + rhalf) * 1028 + col] = acc[rt][j][v] + bias;
    }
    __syncthreads();

    // ===================== scores = q . k^T  (per head, T padded to 32) ====
    // 16 waves <-> 16 tiles: (head, t-tile, row-tile)
    {
        const int h  = wave >> 2;
        const int t0 = ((wave >> 1) & 1) * 16;
        const int rb = (wave & 1) * 16;
        v8f s = {0.f,0.f,0.f,0.f,0.f,0.f,0.f,0.f};
        for (int ks = 0; ks < HD; ks += 4) {
            const float2 a2 =
                *(const float2*)&qtile[(rb + ln) * 1028 + h * HD + ks + koff];
            v2f af = {a2.x, a2.y};
            const float2 b2 =
                *(const float2*)&kbuf[(size_t)(h * TP + t0 + ln) * HD + ks + koff];
            v2f bf = {b2.x, b2.y};
            s = wmma_f32(af, bf, s);
        }
        const float scale = 0.0625f;  // 1/sqrt(256)
        const int t = t0 + ln;
        const float ninf = -__builtin_inff();
#pragma unroll
        for (int v = 0; v < 8; ++v) {
            int row = rb + v + rhalf;   // 0..31
            float val;
            if (t < TT) {
                int m = hour_mask[(size_t)(r0 + row) * TT + t];
                val = (m == 1) ? ninf : s[v] * scale;
            } else {
                val = ninf;
            }
            sc[(h * 32 + row) * TP + t] = val;
        }
    }
    __syncthreads();

    // ===================== softmax over T (one thread per head-row) ========
    if (tid < 128) {
        int h = tid >> 5, row = tid & 31;
        float* p = &sc[(h * 32 + row) * TP];
        float m = -__builtin_inff();
#pragma unroll
        for (int t = 0; t < TT; ++t) m = fmaxf(m, p[t]);
        float sum = 0.f;
#pragma unroll
        for (int t = 0; t < TT; ++t) sum += __expf(p[t] - m);
        float inv = 1.f / sum;
#pragma unroll
        for (int t = 0; t < TT; ++t) p[t] = __expf(p[t] - m) * inv;
#pragma unroll
        for (int t = TT; t < TP; ++t) p[t] = 0.f;
    }
    __syncthreads();

    // ===================== attn . v  -> otile ==============================
    v8f ao[2][4];
#pragma unroll
    for (int rt = 0; rt < 2; ++rt)
#pragma unroll
        for (int j = 0; j < 4; ++j)
            ao[rt][j] = {0.f,0.f,0.f,0.f,0.f,0.f,0.f,0.f};
    {
        const int h  = wave >> 2;             // cb/256
        const int ho = (wave & 3) * 64;       // column offset within head
#pragma unroll
        for (int ks = 0; ks < TP; ks += 4) {
            const float2 a0 =
                *(const float2*)&sc[(h * 32 + ln) * TP + ks + koff];
            const float2 a1 =
                *(const float2*)&sc[(h * 32 + 16 + ln) * TP + ks + koff];
            v2f af0 = {a0.x, a0.y};
            v2f af1 = {a1.x, a1.y};
#pragma unroll
            for (int j = 0; j < 4; ++j) {
                int colh = ho + j * 16 + ln;
                float b0 = vbuf[(size_t)(h * TP + ks + koff) * HD + colh];
                float b1 = vbuf[(size_t)(h * TP + ks + koff + 1) * HD + colh];
                v2f bf = {b0, b1};
                ao[0][j] = wmma_f32(af0, bf, ao[0][j]);
                ao[1][j] = wmma_f32(af1, bf, ao[1][j]);
            }
        }
    }
    __syncthreads();   // all qtile reads done; safe to overwrite as otile
    // global column = (wave>>2)*256 + ho + j*16 + ln == cb + j*16 + ln
#pragma unroll
    for (int j = 0; j < 4; ++j) {
        int col = cb + j * 16 + ln;
#pragma unroll
        for (int rt = 0; rt < 2; ++rt)
#pragma unroll
            for (int v = 0; v < 8; ++v)
                qtile[(rt * 16 + v + rhalf) * 1028 + col] = ao[rt][j][v];
    }
    __syncthreads();

    // ===================== GEMM2: out = otile[32x1024] x Wu^T + bu =========
    v8f accf[2][4];
#pragma unroll
    for (int rt = 0; rt < 2; ++rt)
#pragma unroll
        for (int j = 0; j < 4; ++j)
            accf[rt][j] = {0.f,0.f,0.f,0.f,0.f,0.f,0.f,0.f};
    for (int ks = 0; ks < DM; ks += 4) {
        const float2 a0 = *(const float2*)&qtile[ln * 1028 + ks + koff];
        const float2 a1 = *(const float2*)&qtile[(16 + ln) * 1028 + ks + koff];
        v2f af0 = {a0.x, a0.y};
        v2f af1 = {a1.x, a1.y};
#pragma unroll
        for (int j = 0; j < 4; ++j) {
            int col = cb + j * 16 + ln;
            const float2 b2 = *(const float2*)&Wu[(size_t)col * DM + ks + koff];
            v2f bf = {b2.x, b2.y};
            accf[0][j] = wmma_f32(af0, bf, accf[0][j]);
            accf[1][j] = wmma_f32(af1, bf, accf[1][j]);
        }
    }
#pragma unroll
    for (int j = 0; j < 4; ++j) {
        int col = cb + j * 16 + ln;
        float bias = bu[col];
#pragma unroll
        for (int rt = 0; rt < 2; ++rt)
#pragma unroll
            for (int v = 0; v < 8; ++v)
                out[(size_t)(r0 + rt * 16 + v + rhalf) * DM + col]
                    = accf[rt][j][v] + bias;
    }
}

// ---------------------------------------------------------------------------
extern "C" void kernel_launch(void* const* d_in, const int* in_sizes, int n_in,
                              void* d_out, int out_size, void* d_ws, size_t ws_size,
                              hipStream_t stream)
{
    const float* emb       = (const float*)d_in[0];
    const int*   user_x1   = (const int*)  d_in[1];
    const int*   hour_x1   = (const int*)  d_in[2];
    /* d_in[3] = timeslot_y1 : unused by reference */
    const int*   hour_mask = (const int*)  d_in[4];
    const float* up        = (const float*)d_in[5];
    const float* Wq        = (const float*)d_in[6];
    const float* bq        = (const float*)d_in[7];
    const float* Wk        = (const float*)d_in[8];
    const float* bk        = (const float*)d_in[9];
    const float* Wv        = (const float*)d_in[10];
    const float* bv        = (const float*)d_in[11];
    const float* Wu        = (const float*)d_in[12];
    const float* bu        = (const float*)d_in[13];
    float* out  = (float*)d_out;

    float* kbuf = (float*)d_ws;                 // [NH][TP][HD] = 128 KB
    float* vbuf = kbuf + NH * TP * HD;          // [NH][TP][HD] = 128 KB

    hipLaunchKernelGGL(kv_kernel, dim3(NH * TP), dim3(256), 0, stream,
                       emb, Wk, bk, Wv, bv, kbuf, vbuf);

    const int nblocks = (64 * 512) / BROWS;     // 1024
    hipLaunchKernelGGL(fused_kernel, dim3(nblocks), dim3(NTHREADS), 0, stream,
                       emb, user_x1, hour_x1, hour_mask, up, Wq, bq,
                       kbuf, vbuf, Wu, bu, out);
}